// O_SVGD_68015102100168
// MI455X (gfx1250) — compile-verified
//
#include <hip/hip_runtime.h>

typedef __attribute__((ext_vector_type(2))) float v2f;
typedef __attribute__((ext_vector_type(8))) float v8f;

#define WD_C    0.0005f
#define ALPHA_C 1.0f
#define LR_C    1e-3f

__device__ __forceinline__ v8f wmma4(v2f a, v2f b, v8f c) {
  // D = A(16x4,f32) * B(4x16,f32) + C ; full fp32 matrix-pipe op
  return __builtin_amdgcn_wmma_f32_16x16x4_f32(false, a, false, b, (short)0, c,
                                               false, false);
}

// ---------------------------------------------------------------------------
// Phase 0: zero the 3x(32x32) Gram accumulators (atomics accumulate -> must
// re-zero every launch; harness poisons ws only once).
// ---------------------------------------------------------------------------
__global__ void init_kernel(float* __restrict__ ws) {
  int t = blockIdx.x * blockDim.x + threadIdx.x;
  if (t < 3072) ws[t] = 0.0f;
}

// ---------------------------------------------------------------------------
// Phase 1: S_XX = X X^T, S_XR = X R^T, S_RR = R R^T  (32x32 each, K = D)
// Each wave owns 12 independent 16x16 f32 WMMA accumulator tiles.
// ---------------------------------------------------------------------------
__global__ void __launch_bounds__(256) gram_kernel(const float* __restrict__ X,
                                                   const float* __restrict__ R,
                                                   float* __restrict__ acc,
                                                   long long D) {
  const int lane  = threadIdx.x & 31;
  const int wave  = threadIdx.x >> 5;
  const int gWave = blockIdx.x * (blockDim.x >> 5) + wave;
  const int nWave = gridDim.x * (blockDim.x >> 5);

  const int rA = lane & 15;          // row within 16-row block
  const int cO = (lane >> 4) * 2;    // k-offset within 4-wide slice

  v8f xx00 = {}, xx01 = {}, xx10 = {}, xx11 = {};
  v8f xg00 = {}, xg01 = {}, xg10 = {}, xg11 = {};
  v8f gg00 = {}, gg01 = {}, gg10 = {}, gg11 = {};

  const long long slices = D >> 2;
  for (long long s = gWave; s < slices; s += nWave) {
    const long long k = s * 4 + cO;
    // A-frag and B-frag of a 16-row block share this per-lane layout.
    v2f a0 = *(const v2f*)(X + (long long)rA * D + k);
    v2f a1 = *(const v2f*)(X + (long long)(rA + 16) * D + k);
    v2f g0 = *(const v2f*)(R + (long long)rA * D + k);
    v2f g1 = *(const v2f*)(R + (long long)(rA + 16) * D + k);

    xx00 = wmma4(a0, a0, xx00);  xx01 = wmma4(a0, a1, xx01);
    xx10 = wmma4(a1, a0, xx10);  xx11 = wmma4(a1, a1, xx11);
    xg00 = wmma4(a0, g0, xg00);  xg01 = wmma4(a0, g1, xg01);
    xg10 = wmma4(a1, g0, xg10);  xg11 = wmma4(a1, g1, xg11);
    gg00 = wmma4(g0, g0, gg00);  gg01 = wmma4(g0, g1, gg01);
    gg10 = wmma4(g1, g0, gg10);  gg11 = wmma4(g1, g1, gg11);
  }

  // Cross-wave reduction: 4 LDS regions (48KB). Waves 0-3 store, waves 4-7
  // read-modify-write the same region (one wave per region -> race-free).
  __shared__ float sAcc[4 * 3072];
  float* w = sAcc + (wave & 3) * 3072;
  const int colN = lane & 15;
  const int mB   = (lane >> 4) * 8;
  const bool first = (wave < 4);

  if (first) {
#pragma unroll
    for (int v = 0; v < 8; ++v) {
      int m0 = mB + v, m1 = m0 + 16, c1 = colN + 16;
      w[0 * 1024 + m0 * 32 + colN] = xx00[v];
      w[0 * 1024 + m0 * 32 + c1  ] = xx01[v];
      w[0 * 1024 + m1 * 32 + colN] = xx10[v];
      w[0 * 1024 + m1 * 32 + c1  ] = xx11[v];
      w[1 * 1024 + m0 * 32 + colN] = xg00[v];
      w[1 * 1024 + m0 * 32 + c1  ] = xg01[v];
      w[1 * 1024 + m1 * 32 + colN] = xg10[v];
      w[1 * 1024 + m1 * 32 + c1  ] = xg11[v];
      w[2 * 1024 + m0 * 32 + colN] = gg00[v];
      w[2 * 1024 + m0 * 32 + c1  ] = gg01[v];
      w[2 * 1024 + m1 * 32 + colN] = gg10[v];
      w[2 * 1024 + m1 * 32 + c1  ] = gg11[v];
    }
  }
  __syncthreads();
  if (!first) {
#pragma unroll
    for (int v = 0; v < 8; ++v) {
      int m0 = mB + v, m1 = m0 + 16, c1 = colN + 16;
      w[0 * 1024 + m0 * 32 + colN] += xx00[v];
      w[0 * 1024 + m0 * 32 + c1  ] += xx01[v];
      w[0 * 1024 + m1 * 32 + colN] += xx10[v];
      w[0 * 1024 + m1 * 32 + c1  ] += xx11[v];
      w[1 * 1024 + m0 * 32 + colN] += xg00[v];
      w[1 * 1024 + m0 * 32 + c1  ] += xg01[v];
      w[1 * 1024 + m1 * 32 + colN] += xg10[v];
      w[1 * 1024 + m1 * 32 + c1  ] += xg11[v];
      w[2 * 1024 + m0 * 32 + colN] += gg00[v];
      w[2 * 1024 + m0 * 32 + c1  ] += gg01[v];
      w[2 * 1024 + m1 * 32 + colN] += gg10[v];
      w[2 * 1024 + m1 * 32 + c1  ] += gg11[v];
    }
  }
  __syncthreads();
  for (int e = threadIdx.x; e < 3072; e += blockDim.x) {
    float s = sAcc[e] + sAcc[3072 + e] + sAcc[2 * 3072 + e] + sAcc[3 * 3072 + e];
    atomicAdd(&acc[e], s);
  }
}

// ---------------------------------------------------------------------------
// Phase 2: all 32x32 algebra. One workgroup, thread t = (i,j) = (t/32, t%32).
// Produces Pf, Qf such that out = X + clip(LR*(Pf@X + Qf@R), +-1000).
// ---------------------------------------------------------------------------
__global__ void __launch_bounds__(1024) coeff_kernel(const float* __restrict__ gram,
                                                     const float* __restrict__ gloss,
                                                     const int* __restrict__ epochP,
                                                     float* __restrict__ pq) {
  __shared__ float sXX[1024], sXR[1024], sRR[1024];
  __shared__ float sXG[1024], sGG[1024], sPD[1024];
  __shared__ float sCv[1024], sAv[1024], sKv[1024];
  __shared__ float sMed[2];

  const int t = threadIdx.x;
  const int i = t >> 5, j = t & 31;
  const int  epoch = *epochP;
  const bool late  = (epoch >= 150);

  sXX[t] = gram[t];
  sXR[t] = gram[1024 + t];
  sRR[t] = gram[2048 + t];
  if (t < 2) sMed[t] = 0.0f;
  __syncthreads();

  // Grams in G-space (G = R for late epochs, G = R + WD*X otherwise)
  float xg, gg;
  if (late) {
    xg = sXR[t];
    gg = sRR[t];
  } else {
    float xrT = sXR[j * 32 + i];
    xg = sXR[t] + WD_C * sXX[t];
    gg = sRR[t] + WD_C * (sXR[t] + xrT) + WD_C * WD_C * sXX[t];
  }
  sXG[t] = xg;
  sGG[t] = gg;

  float pd = sXX[i * 33] + sXX[j * 33] - 2.0f * sXX[t];
  pd = fmaxf(pd, 0.0f);
  sPD[t] = pd;
  __syncthreads();

  // Exact median of 1024 values (mean of order stats 511 and 512) via ranks.
  int less = 0, leq = 0;
  for (int m = 0; m < 1024; ++m) {
    float v = sPD[m];
    less += (v < pd) ? 1 : 0;
    leq  += (v <= pd) ? 1 : 0;
  }
  if (less <= 511 && 511 < leq) sMed[0] = pd;   // same-value races are benign
  if (less <= 512 && 512 < leq) sMed[1] = pd;
  __syncthreads();

  const float hk = (sMed[0] + sMed[1]) * 0.5f / logf(32.0f) + 1e-6f;

  const float Kv = expf(-pd / hk);
  const float Cv = 2.0f * Kv / hk;

  const float dXG_i = sXG[i * 33], dXG_j = sXG[j * 33];
  const float Gn2_i = sGG[i * 33], Gn2_j = sGG[j * 33];

  const float Av = Cv * (dXG_i - sXG[j * 32 + i]) / Gn2_i;
  const float Bv = (Cv * (sXG[t] - dXG_j) - Av * sGG[t]) / Gn2_j;

  float Mv = 0.0f, omc_i = 0.0f;                 // omc = 1 - coef
  if (late) {
    omc_i = -WD_C * dXG_i / Gn2_i;
    float sgt = WD_C * sXG[t] + omc_i * sGG[t];  // (S @ G^T)_ij
    Mv = Kv * sgt / Gn2_j;
  }

  sCv[t] = Cv; sAv[t] = Av; sKv[t] = Kv;
  __syncthreads();

  float rsC = 0.0f, rsA = 0.0f, rsK = 0.0f;
  for (int m = 0; m < 32; ++m) {
    rsC += sCv[i * 32 + m];
    rsA += sAv[i * 32 + m];
    rsK += sKv[i * 32 + m];
  }

  const float delta = (i == j) ? 1.0f : 0.0f;
  float Pg = delta * (rsC + (late ? WD_C * rsK : 0.0f)) - Cv;
  float Qg = delta * ((late ? rsK * omc_i : 0.0f) - rsA) - Bv - Mv;

  float g   = gloss[i];
  float sgn = (g > 0.0f ? 1.0f : 0.0f) - (g < 0.0f ? 1.0f : 0.0f);
  float Qf  = Qg * (1.0f / 32.0f) - delta * ALPHA_C * sgn;
  float Pf  = Pg * (1.0f / 32.0f) + (late ? 0.0f : WD_C * Qf);

  pq[t]        = Pf;
  pq[1024 + t] = Qf;
}

// ---------------------------------------------------------------------------
// Phase 3: out = X + clip(LR*(Pf@X + Qf@R)). Pf/Qf A-fragments loaded once
// per wave; each wave streams 16-column slabs with 32 WMMAs per slab.
// ---------------------------------------------------------------------------
__global__ void __launch_bounds__(256) apply_kernel(const float* __restrict__ X,
                                                    const float* __restrict__ R,
                                                    const float* __restrict__ PQ,
                                                    float* __restrict__ OUT,
                                                    long long D) {
  const int lane  = threadIdx.x & 31;
  const int wave  = threadIdx.x >> 5;
  const int gWave = blockIdx.x * (blockDim.x >> 5) + wave;
  const int nWave = gridDim.x * (blockDim.x >> 5);

  const int rA = lane & 15;
  const int cO = (lane >> 4) * 2;

  v2f pf0[8], pf1[8], qf0[8], qf1[8];
#pragma unroll
  for (int ks = 0; ks < 8; ++ks) {
    int kc = ks * 4 + cO;
    pf0[ks] = *(const v2f*)(PQ + rA * 32 + kc);
    pf1[ks] = *(const v2f*)(PQ + (rA + 16) * 32 + kc);
    qf0[ks] = *(const v2f*)(PQ + 1024 + rA * 32 + kc);
    qf1[ks] = *(const v2f*)(PQ + 1024 + (rA + 16) * 32 + kc);
  }

  const long long slabs = D >> 4;
  for (long long sl = gWave; sl < slabs; sl += nWave) {
    const long long col = sl * 16 + (lane & 15);

    v2f bx[8], bg[8];
#pragma unroll
    for (int ks = 0; ks < 8; ++ks) {
      int r0 = ks * 4 + cO;
      bx[ks][0] = X[(long long)r0 * D + col];
      bx[ks][1] = X[(long long)(r0 + 1) * D + col];
      bg[ks][0] = R[(long long)r0 * D + col];
      bg[ks][1] = R[(long long)(r0 + 1) * D + col];
    }

    v8f acc0 = {}, acc1 = {};
#pragma unroll
    for (int ks = 0; ks < 8; ++ks) {
      acc0 = wmma4(pf0[ks], bx[ks], acc0);
      acc0 = wmma4(qf0[ks], bg[ks], acc0);
      acc1 = wmma4(pf1[ks], bx[ks], acc1);
      acc1 = wmma4(qf1[ks], bg[ks], acc1);
    }

    const int mB = (lane >> 4) * 8;
#pragma unroll
    for (int v = 0; v < 8; ++v) {
      int m = mB + v;
      float u0 = fminf(fmaxf(LR_C * acc0[v], -1000.0f), 1000.0f);
      float u1 = fminf(fmaxf(LR_C * acc1[v], -1000.0f), 1000.0f);
      OUT[(long long)m * D + col]        = X[(long long)m * D + col] + u0;
      OUT[(long long)(m + 16) * D + col] = X[(long long)(m + 16) * D + col] + u1;
    }
  }
}

// ---------------------------------------------------------------------------
extern "C" void kernel_launch(void* const* d_in, const int* in_sizes, int n_in,
                              void* d_out, int out_size, void* d_ws, size_t ws_size,
                              hipStream_t stream) {
  const float* X     = (const float*)d_in[0];   // particles [32, D]
  const float* R     = (const float*)d_in[1];   // raw_grad  [32, D]
  const float* gloss = (const float*)d_in[2];   // [32]
  const int*   epoch = (const int*)d_in[3];     // scalar
  float* OUT = (float*)d_out;
  float* ws  = (float*)d_ws;
  const long long D = (long long)in_sizes[0] / 32;

  hipLaunchKernelGGL(init_kernel,  dim3(12),   dim3(256),  0, stream, ws);
  hipLaunchKernelGGL(gram_kernel,  dim3(128),  dim3(256),  0, stream, X, R, ws, D);
  hipLaunchKernelGGL(coeff_kernel, dim3(1),    dim3(1024), 0, stream, ws, gloss, epoch, ws + 3072);
  hipLaunchKernelGGL(apply_kernel, dim3(1024), dim3(256),  0, stream, X, R, ws + 3072, OUT, D);
}